// LIFNode_61881888800873
// MI455X (gfx1250) — compile-verified
//
#include <hip/hip_runtime.h>

typedef __attribute__((ext_vector_type(4))) float v4f;

#define T_DIM   128
#define D_DIM   32768
#define DECAY_F 0.5f      // 1 - 1/tau, tau = 2
#define V_TH_F  1.0f

#define BLOCK       256
#define VEC         4
#define DEPTH       16                 // async prefetch depth (power of 2)
#define WAVES       (BLOCK / 32)
#define STAGE_BYTES (32 * 16)          // 32 lanes * 16B per stage per wave

__device__ __forceinline__ void lif_step(float& v, float xin, float& spk) {
    v   = __builtin_fmaf(v, DECAY_F, xin);   // leaky integrate
    spk = (v >= V_TH_F) ? 1.0f : 0.0f;       // heaviside fire
    v  -= spk;                               // soft reset (V_TH = 1)
}

__global__ __launch_bounds__(BLOCK, 1) void lif_fwd_kernel(
    const float* __restrict__ x, float* __restrict__ out) {
    // Per-wave private ring buffer: no cross-wave sharing, no barriers needed.
    __shared__ v4f sbuf[WAVES * DEPTH * 32];

    const int lane = threadIdx.x & 31;
    const int wave = threadIdx.x >> 5;
    const int n    = blockIdx.y;
    const long long d = (long long)blockIdx.x * (BLOCK * VEC) + (long long)threadIdx.x * VEC;

    const long long nbase = (long long)n * T_DIM * D_DIM + d;
    const unsigned long long gbase   = (unsigned long long)x + (unsigned long long)nbase * 4ull;
    const unsigned long long tstride = (unsigned long long)D_DIM * 4ull;   // bytes per t step

    const int      slot0 = (wave * DEPTH) * 32 + lane;
    // Low 32 bits of the generic pointer to LDS == wave-relative LDS byte offset.
    const unsigned lbase = (unsigned)(unsigned long long)(&sbuf[slot0]);

    // ---- prologue: fill the async ring (DEPTH stages in flight, held in LDS not VGPRs)
#pragma unroll
    for (int k = 0; k < DEPTH; ++k) {
        const unsigned           laddr = lbase + (unsigned)(k * STAGE_BYTES);
        const unsigned long long gaddr = gbase + (unsigned long long)k * tstride;
        asm volatile("global_load_async_to_lds_b128 %0, %1, off"
                     :: "v"(laddr), "v"(gaddr) : "memory");
    }

    float v0 = 0.f, v1 = 0.f, v2 = 0.f, v3 = 0.f;
    float* const outp = out + nbase;

    // ---- steady state: consume stage t, refill it with t+DEPTH
    for (int t = 0; t < T_DIM - DEPTH; ++t) {
        const int s = t & (DEPTH - 1);
        // async loads complete in order -> <=DEPTH-1 outstanding means stage t landed
        asm volatile("s_wait_asynccnt %0" :: "n"(DEPTH - 1) : "memory");
        v4f xv = sbuf[slot0 + s * 32];                 // ds_load_b128
        // make sure the LDS read retired before the async write reuses this stage
        asm volatile("s_wait_dscnt 0" ::: "memory");
        {
            const unsigned           laddr = lbase + (unsigned)(s * STAGE_BYTES);
            const unsigned long long gaddr = gbase + (unsigned long long)(t + DEPTH) * tstride;
            asm volatile("global_load_async_to_lds_b128 %0, %1, off"
                         :: "v"(laddr), "v"(gaddr) : "memory");
        }
        float s0, s1, s2, s3;
        lif_step(v0, xv.x, s0);
        lif_step(v1, xv.y, s1);
        lif_step(v2, xv.z, s2);
        lif_step(v3, xv.w, s3);
        v4f sp;
        sp.x = s0; sp.y = s1; sp.z = s2; sp.w = s3;
        // streaming output, never re-read: non-temporal store keeps L2 for the input stream
        __builtin_nontemporal_store(sp, (v4f*)(outp + (long long)t * D_DIM));
    }

    // ---- epilogue: drain ring, no further prefetch
    asm volatile("s_wait_asynccnt 0" ::: "memory");
#pragma unroll
    for (int k = 0; k < DEPTH; ++k) {
        const int t = T_DIM - DEPTH + k;
        const int s = t & (DEPTH - 1);
        v4f xv = sbuf[slot0 + s * 32];
        float s0, s1, s2, s3;
        lif_step(v0, xv.x, s0);
        lif_step(v1, xv.y, s1);
        lif_step(v2, xv.z, s2);
        lif_step(v3, xv.w, s3);
        v4f sp;
        sp.x = s0; sp.y = s1; sp.z = s2; sp.w = s3;
        __builtin_nontemporal_store(sp, (v4f*)(outp + (long long)t * D_DIM));
    }
}

extern "C" void kernel_launch(void* const* d_in, const int* in_sizes, int n_in,
                              void* d_out, int out_size, void* d_ws, size_t ws_size,
                              hipStream_t stream) {
    const float* x   = (const float*)d_in[0];
    float*       out = (float*)d_out;

    const int total = in_sizes[0];            // N * T * D
    const int n     = total / (T_DIM * D_DIM);  // 16 for the reference shapes

    dim3 grid(D_DIM / (BLOCK * VEC), n);      // (32, 16)
    dim3 block(BLOCK);
    lif_fwd_kernel<<<grid, block, 0, stream>>>(x, out);
}